// LatentTimeStreamLayer_45724221833762
// MI455X (gfx1250) — compile-verified
//
#include <hip/hip_runtime.h>

typedef __attribute__((ext_vector_type(2))) float v2f;
typedef __attribute__((ext_vector_type(4))) float v4f;
typedef __attribute__((ext_vector_type(8))) float v8f;

// Problem dims from the reference (fixed).
constexpr int B  = 128;
constexpr int T  = 512;
constexpr int DZ = 512;

// ---------------------------------------------------------------------------
// Kernel 1: exact f32 inclusive prefix-sum of x[b, :] along T.
// One block per batch row, T threads, Hillis-Steele scan in LDS.
// Total data: 256 KB in / 256 KB out -- negligible vs the 134 MB main stream.
// ---------------------------------------------------------------------------
__global__ void scan_kernel(const float* __restrict__ x, float* __restrict__ S) {
  __shared__ float buf[T];
  const int b = blockIdx.x;
  const int t = threadIdx.x;

  float v = x[(size_t)b * T + t];
  buf[t] = v;
  __syncthreads();

  for (int off = 1; off < T; off <<= 1) {
    float add = (t >= off) ? buf[t - off] : 0.0f;
    __syncthreads();
    v += add;
    buf[t] = v;
    __syncthreads();
  }
  S[(size_t)b * T + t] = v;
}

// ---------------------------------------------------------------------------
// Kernel 2: out[b,t,d] = S[b,t]*f[d] + z0[b,d] via V_WMMA_F32_16X16X4_F32.
// Tile orientation: M = d, N = t.
//   A (16x4 f32): A[d,0] = f[d],   K=1..3 zero  -> a0 = (lane<16)?f:0, a1 = 0
//   B (4x16 f32): B[0,t] = S[t],   K=1..3 zero  -> b0 = (lane<16)?S:0, b1 = 0
//   C (16x16 f32): C[d,t] = z0[b,d]  (broadcast over t / lanes)
// D layout (f32 16x16): VGPR i, lanes 0-15 -> (M=i, N=lane);
//                       lanes 16-31 -> (M=i+8, N=lane-16).
// => each lane stores d-contiguous runs: two b128 stores per lane.
// ---------------------------------------------------------------------------
__global__ void rank1_wmma_kernel(const float* __restrict__ S,
                                  const float* __restrict__ z0,
                                  const float* __restrict__ f,
                                  float* __restrict__ out) {
  constexpr int DT = DZ / 16;  // 32 d-tiles
  constexpr int TT = T / 16;   // 32 t-tiles

  const int wave = blockIdx.x * (blockDim.x >> 5) + (threadIdx.x >> 5);
  const int lane = threadIdx.x & 31;

  const int dt = wave & (DT - 1);          // d tile
  const int tt = (wave >> 5) & (TT - 1);   // t tile
  const int b  = wave >> 10;               // batch

  const bool lo  = lane < 16;
  const int  l16 = lane & 15;

  // A: f[d] in K=0 column.
  v2f a;
  a[0] = lo ? f[dt * 16 + l16] : 0.0f;
  a[1] = 0.0f;

  // B: S[t] in K=0 row.
  v2f bm;
  bm[0] = lo ? S[(size_t)b * T + tt * 16 + l16] : 0.0f;
  bm[1] = 0.0f;

  // C: z0[b, d] broadcast across the 16 t-columns of each lane half.
  const float* z0row = z0 + (size_t)b * DZ + dt * 16 + (lo ? 0 : 8);
  v8f c;
  c[0] = z0row[0]; c[1] = z0row[1]; c[2] = z0row[2]; c[3] = z0row[3];
  c[4] = z0row[4]; c[5] = z0row[5]; c[6] = z0row[6]; c[7] = z0row[7];

  // D = A x B + C  (f32 in, f32 accumulate: bitwise-equivalent precision path)
  v8f d = __builtin_amdgcn_wmma_f32_16x16x4_f32(
      /*neg_a=*/false, a, /*neg_b=*/false, bm,
      /*c_mod=*/(short)0, c, /*reuse_a=*/false, /*reuse_b=*/false);

  // Store: lane owns t = tt*16 + l16, d = dt*16 + (lo?0:8) + i for i=0..7.
  const size_t base = ((size_t)b * T + tt * 16 + l16) * DZ + dt * 16 + (lo ? 0 : 8);
  v4f d03 = {d[0], d[1], d[2], d[3]};
  v4f d47 = {d[4], d[5], d[6], d[7]};
  *(v4f*)(out + base)     = d03;   // 16B-aligned (offsets are multiples of 8 floats)
  *(v4f*)(out + base + 4) = d47;
}

// ---------------------------------------------------------------------------
// Launch: scan into d_ws (needs B*T*4 = 256 KB), then WMMA rank-1 stream.
// ---------------------------------------------------------------------------
extern "C" void kernel_launch(void* const* d_in, const int* in_sizes, int n_in,
                              void* d_out, int out_size, void* d_ws, size_t ws_size,
                              hipStream_t stream) {
  const float* x  = (const float*)d_in[0];  // t_inputs  (B, T, 1)
  const float* z0 = (const float*)d_in[1];  // init_states (B, DZ)
  const float* f  = (const float*)d_in[2];  // f (1, DZ)
  float* out = (float*)d_out;
  float* S   = (float*)d_ws;                // B*T floats of scratch

  (void)in_sizes; (void)n_in; (void)out_size; (void)ws_size;

  scan_kernel<<<B, T, 0, stream>>>(x, S);

  constexpr int totalWaves = B * (T / 16) * (DZ / 16);  // 131072 tiles
  constexpr int blockThreads = 256;                     // 8 waves/block
  constexpr int grid = totalWaves / (blockThreads / 32);
  rank1_wmma_kernel<<<grid, blockThreads, 0, stream>>>(S, z0, f, out);
}